// GCN_54906861912516
// MI455X (gfx1250) — compile-verified
//
#include <hip/hip_runtime.h>
#include <math.h>

typedef float v2f __attribute__((ext_vector_type(2)));
typedef float v8f __attribute__((ext_vector_type(8)));

// ---------------------------------------------------------------------------
// Degree / normalization kernels
// ---------------------------------------------------------------------------
__global__ void k_init_deg(float* __restrict__ deg, int n) {
    int i = blockIdx.x * blockDim.x + threadIdx.x;
    if (i < n) deg[i] = 1.0f;  // self-loop contributes 1 to every node's degree
}

__global__ void k_deg_edges(const int* __restrict__ col, float* __restrict__ deg, int ne) {
    int e = blockIdx.x * blockDim.x + threadIdx.x;
    if (e < ne) atomicAdd(&deg[col[e]], 1.0f);
}

__global__ void k_deg_to_dis(float* __restrict__ d, int n) {
    int i = blockIdx.x * blockDim.x + threadIdx.x;
    if (i < n) d[i] = rsqrtf(d[i]);   // deg >= 1 always (self-loop)
}

// ---------------------------------------------------------------------------
// WMMA fp32 GEMM: Y[nrows x NOUT] = (RELU_IN ? relu(X) : X)[nrows x K] @ W[K x NOUT]
// One wave computes one 16x16 output tile using V_WMMA_F32_16X16X4_F32.
// ---------------------------------------------------------------------------
template <int K, int NOUT, bool RELU_IN>
__global__ void k_gemm_wmma(const float* __restrict__ X, const float* __restrict__ W,
                            float* __restrict__ Y, int nrows) {
    constexpr int TILES_N = (NOUT + 15) / 16;
    const int wave = (int)((blockIdx.x * blockDim.x + threadIdx.x) >> 5);
    const int lane = threadIdx.x & 31;
    const int total_tiles = (nrows / 16) * TILES_N;
    if (wave >= total_tiles) return;              // uniform per-wave exit: EXEC stays all-1s

    const int tm   = wave / TILES_N;
    const int tn   = wave % TILES_N;
    const int m0   = tm * 16;
    const int n0   = tn * 16;
    const int half = lane >> 4;                   // 0: K pair {0,1}, 1: K pair {2,3}
    const int r    = lane & 15;
    const int n    = n0 + r;
    const bool col_ok = (n < NOUT);               // zero-pad for NOUT=4 (layer 3)

    const float* __restrict__ xrow = X + (size_t)(m0 + r) * K;

    v8f acc = {};
    for (int k = 0; k < K; k += 4) {
        const int ka = k + 2 * half;
        float2 av = *(const float2*)(xrow + ka);  // contiguous pair -> global_load_b64
        v2f a;
        a[0] = RELU_IN ? fmaxf(av.x, 0.0f) : av.x;
        a[1] = RELU_IN ? fmaxf(av.y, 0.0f) : av.y;
        v2f b;
        b[0] = col_ok ? W[(size_t)ka * NOUT + n]       : 0.0f;
        b[1] = col_ok ? W[(size_t)(ka + 1) * NOUT + n] : 0.0f;
        // D = A(16x4,f32) * B(4x16,f32) + C ; full-precision accumulate
        acc = __builtin_amdgcn_wmma_f32_16x16x4_f32(
            /*neg_a=*/false, a, /*neg_b=*/false, b,
            /*c_mod=*/(short)0, acc, /*reuse_a=*/false, /*reuse_b=*/false);
    }

    // D layout: VGPR v -> row m0 + v + 8*half, col n0 + r
    if (col_ok) {
#pragma unroll
        for (int v = 0; v < 8; ++v) {
            const int m = m0 + v + 8 * half;
            Y[(size_t)m * NOUT + n] = acc[v];
        }
    }
}

// ---------------------------------------------------------------------------
// Output init: fuse bias + self-loop contribution (norm = dis[i]^2)
//   out[i, f] = b[f] + H[i, f] * dis[i]^2
// ---------------------------------------------------------------------------
template <int F>
__global__ void k_init_out(const float* __restrict__ H, const float* __restrict__ bias,
                           const float* __restrict__ dis, float* __restrict__ out, int n) {
    long long tid = (long long)blockIdx.x * blockDim.x + threadIdx.x;
    if (tid >= (long long)n * F) return;
    const int i = (int)(tid / F);
    const int f = (int)(tid % F);
    const float d = dis[i];
    out[tid] = bias[f] + H[tid] * d * d;
}

// ---------------------------------------------------------------------------
// Edge scatter: out[col[e], :] += H[row[e], :] * dis[row]*dis[col]
// F/4 lanes per edge; float4 gathers + 4 global_atomic_add_f32 per lane.
// ---------------------------------------------------------------------------
template <int F>
__global__ void k_scatter_edges(const float* __restrict__ H, const int* __restrict__ row,
                                const int* __restrict__ col, const float* __restrict__ dis,
                                float* __restrict__ out, int ne) {
    constexpr int TPE = F / 4;  // lanes per edge (16 for F=64, 1 for F=4)
    long long tid = (long long)blockIdx.x * blockDim.x + threadIdx.x;
    const int e = (int)(tid / TPE);
    const int q = (int)(tid % TPE);
    if (e >= ne) return;
    const int r = row[e];
    const int c = col[e];
    const float w = dis[r] * dis[c];
    const float4 h = *(const float4*)(H + (size_t)r * F + q * 4);
    float* o = out + (size_t)c * F + q * 4;
    atomicAdd(o + 0, h.x * w);
    atomicAdd(o + 1, h.y * w);
    atomicAdd(o + 2, h.z * w);
    atomicAdd(o + 3, h.w * w);
}

// ---------------------------------------------------------------------------
// Host launcher
// ---------------------------------------------------------------------------
extern "C" void kernel_launch(void* const* d_in, const int* in_sizes, int n_in,
                              void* d_out, int out_size, void* d_ws, size_t ws_size,
                              hipStream_t stream) {
    const float* x   = (const float*)d_in[0];   // [N, 128]
    const int*   ei  = (const int*)d_in[1];     // [2, E]
    const float* W1  = (const float*)d_in[2];   // [128, 64]
    const float* b1  = (const float*)d_in[3];
    const float* W2  = (const float*)d_in[4];   // [64, 64]
    const float* b2  = (const float*)d_in[5];
    const float* W3  = (const float*)d_in[6];   // [64, 4]
    const float* b3  = (const float*)d_in[7];
    float* out = (float*)d_out;                 // [N, 4]

    const int n  = in_sizes[0] / 128;           // 100000 nodes (multiple of 16)
    const int ne = in_sizes[1] / 2;             // 3.2M edges
    const int* row = ei;                        // edge_index[0]
    const int* colv = ei + ne;                  // edge_index[1]

    // Workspace layout: dis [n] | bufA [n*64] | bufB [n*64]
    char* ws = (char*)d_ws;
    auto align256 = [](size_t v) { return (v + 255) & ~(size_t)255; };
    float* dis  = (float*)ws;
    size_t off  = align256((size_t)n * sizeof(float));
    float* bufA = (float*)(ws + off);
    size_t szH  = align256((size_t)n * 64 * sizeof(float));
    float* bufB = (float*)(ws + off + szH);

    const int BLK = 256;
    auto blocks1d = [](long long work, int blk) { return (unsigned)((work + blk - 1) / blk); };

    // --- normalization: deg -> dis (in place) ---
    k_init_deg  <<<blocks1d(n, BLK),  BLK, 0, stream>>>(dis, n);
    k_deg_edges <<<blocks1d(ne, BLK), BLK, 0, stream>>>(colv, dis, ne);
    k_deg_to_dis<<<blocks1d(n, BLK),  BLK, 0, stream>>>(dis, n);

    // --- layer 1: H1 = x @ W1 ; A1 = aggregate(H1) + b1 ---
    {
        const int tiles = (n / 16) * 4;           // NOUT=64 -> 4 n-tiles
        k_gemm_wmma<128, 64, false><<<blocks1d((long long)tiles * 32, BLK), BLK, 0, stream>>>(x, W1, bufA, n);
        k_init_out<64><<<blocks1d((long long)n * 64, BLK), BLK, 0, stream>>>(bufA, b1, dis, bufB, n);
        k_scatter_edges<64><<<blocks1d((long long)ne * 16, BLK), BLK, 0, stream>>>(bufA, row, colv, dis, bufB, ne);
    }

    // --- layer 2: H2 = relu(A1) @ W2 ; A2 = aggregate(H2) + b2 ---
    {
        const int tiles = (n / 16) * 4;
        k_gemm_wmma<64, 64, true><<<blocks1d((long long)tiles * 32, BLK), BLK, 0, stream>>>(bufB, W2, bufA, n);
        k_init_out<64><<<blocks1d((long long)n * 64, BLK), BLK, 0, stream>>>(bufA, b2, dis, bufB, n);
        k_scatter_edges<64><<<blocks1d((long long)ne * 16, BLK), BLK, 0, stream>>>(bufA, row, colv, dis, bufB, ne);
    }

    // --- layer 3: H3 = A2 @ W3 ; out = aggregate(H3) + b3 ---
    {
        const int tiles = (n / 16) * 1;           // NOUT=4 -> 1 zero-padded n-tile
        k_gemm_wmma<64, 4, false><<<blocks1d((long long)tiles * 32, BLK), BLK, 0, stream>>>(bufB, W3, bufA, n);
        k_init_out<4><<<blocks1d((long long)n * 4, BLK), BLK, 0, stream>>>(bufA, b3, dis, out, n);
        k_scatter_edges<4><<<blocks1d((long long)ne, BLK), BLK, 0, stream>>>(bufA, row, colv, dis, out, ne);
    }
}